// Polar_36928128811118
// MI455X (gfx1250) — compile-verified
//
#include <hip/hip_runtime.h>
#include <math.h>

// Problem constants (from the reference: x = [64, 3, 512, 512] fp32).
#define B_  64
#define C_  3
#define H_  512
#define W_  512

// 2*pi / 512
#define PHI_STEP 0.0122718463030851302645f

// Tile: 64 h-values (contiguous output) x 4 w-values, 256 threads = 8 wave32.
#define TILE_H 64
#define TILE_W 4
// Each block reuses its register-resident coordinate map across 8 planes.
#define PLANES_PER_BLOCK 8

__global__ __launch_bounds__(TILE_H * TILE_W) void polar_warp_kernel(
    const float* __restrict__ in, float* __restrict__ out)
{
    const int h = blockIdx.x * TILE_H + threadIdx.x;   // output angle index
    const int w = blockIdx.y * TILE_W + threadIdx.y;   // output radius index

    // ---- Coordinate map: depends only on (h, w); computed ONCE, reused for
    // ---- all PLANES_PER_BLOCK (b,c) planes handled by this block.
    const float phi = (float)h * PHI_STEP;
    const float rho = (float)w * 0.5f;                 // w * (maxR / W), maxR = 256
    float sphi, cphi;
    __sincosf(phi, &sphi, &cphi);
    const float sx = 256.0f + rho * cphi;              // source column (W axis)
    const float sy = 256.0f + rho * sphi;              // source row    (H axis)

    const float x0f = floorf(sx);
    const float y0f = floorf(sy);
    const float tx = sx - x0f;
    const float ty = sy - y0f;
    const int x0 = (int)x0f;
    const int y0 = (int)y0f;
    const int x1 = x0 + 1;
    const int y1 = y0 + 1;

    // Per-tap validity (zero-fill outside source), folded into the weights.
    const float vx0 = (x0 >= 0 && x0 < W_) ? 1.0f : 0.0f;
    const float vx1 = (x1 >= 0 && x1 < W_) ? 1.0f : 0.0f;
    const float vy0 = (y0 >= 0 && y0 < H_) ? 1.0f : 0.0f;
    const float vy1 = (y1 >= 0 && y1 < H_) ? 1.0f : 0.0f;

    const float w00 = (1.0f - tx) * (1.0f - ty) * vx0 * vy0;
    const float w01 = tx          * (1.0f - ty) * vx1 * vy0;
    const float w10 = (1.0f - tx) * ty          * vx0 * vy1;
    const float w11 = tx          * ty          * vx1 * vy1;

    // Clamped offsets so the (weight==0) taps still read in-bounds memory.
    const int x0c = min(max(x0, 0), W_ - 1);
    const int x1c = min(max(x1, 0), W_ - 1);
    const int y0c = min(max(y0, 0), H_ - 1);
    const int y1c = min(max(y1, 0), H_ - 1);

    const int o00 = y0c * W_ + x0c;
    const int o01 = y0c * W_ + x1c;
    const int o10 = y1c * W_ + x0c;
    const int o11 = y1c * W_ + x1c;

    // ---- Plane loop: 4 gathers + 3 FMAs + 1 NT store per plane.
    const size_t planeStride = (size_t)H_ * (size_t)W_;
    const int bcBase = blockIdx.z * PLANES_PER_BLOCK;

    const float* __restrict__ p = in + (size_t)bcBase * planeStride;
    // Output layout [B, C, W, H]: index = ((bc)*W + w)*H + h; per-plane
    // stride is also W_*H_ == planeStride.
    float* __restrict__ q = out + ((size_t)bcBase * (size_t)W_ + (size_t)w) * (size_t)H_
                                + (size_t)h;

    // Warm the first plane's two source rows (gfx1250 global_prefetch_b8;
    // does not increment LOADcnt).
    __builtin_prefetch(p + o00, 0, 3);
    __builtin_prefetch(p + o10, 0, 3);

    #pragma unroll
    for (int i = 0; i < PLANES_PER_BLOCK; ++i) {
        // Prefetch the NEXT plane's full bilinear footprint: the 4 taps span
        // two source rows 2 KB apart, i.e. two distinct cachelines.
        if (i + 1 < PLANES_PER_BLOCK) {
            __builtin_prefetch(p + planeStride + o00, 0, 3);
            __builtin_prefetch(p + planeStride + o10, 0, 3);
        }

        const float v00 = p[o00];
        const float v01 = p[o01];
        const float v10 = p[o10];
        const float v11 = p[o11];

        const float v = fmaf(w00, v00,
                        fmaf(w01, v01,
                        fmaf(w10, v10, w11 * v11)));

        // Streamed, write-once output: non-temporal store keeps the 192 MB L2
        // available for the (re-read) input planes.
        __builtin_nontemporal_store(v, q);

        p += planeStride;
        q += planeStride;
    }
}

extern "C" void kernel_launch(void* const* d_in, const int* in_sizes, int n_in,
                              void* d_out, int out_size, void* d_ws, size_t ws_size,
                              hipStream_t stream) {
    (void)in_sizes; (void)n_in; (void)out_size; (void)d_ws; (void)ws_size;

    const float* x = (const float*)d_in[0];
    float* out = (float*)d_out;

    dim3 block(TILE_H, TILE_W, 1);                              // 256 threads = 8 waves
    dim3 grid(H_ / TILE_H, W_ / TILE_W, (B_ * C_) / PLANES_PER_BLOCK);

    hipLaunchKernelGGL(polar_warp_kernel, grid, block, 0, stream, x, out);
}